// Word2Vec_70076686401740
// MI455X (gfx1250) — compile-verified
//
#include <hip/hip_runtime.h>
#include <hip/hip_bf16.h>
#include <math.h>

// ---------------------------------------------------------------------------
// Word2Vec negative-sampling loss, algebraically collapsed:
// only 6 score entries per batch row are ever consumed, so we compute
// 2048 x 6 length-128 dot products (gathered rows) instead of the
// 2048 x 100000 projection. Dot products run on v_wmma_f32_16x16x4_f32:
// one wave32 per 16-row tile, 6 accumulators (one per sample slot),
// useful results on the 16x16 diagonal.
// ---------------------------------------------------------------------------

typedef __attribute__((ext_vector_type(2))) float v2f;
typedef __attribute__((ext_vector_type(8))) float v8f;

#define W2V_EMBED 128
#define W2V_NEG   5
#define W2V_BATCH 2048
#define W2V_SLOTS 6            // 1 positive + 5 negatives
#define ROWS_PER_BLOCK 16      // one 16x16 WMMA tile of batch rows per wave

__device__ __forceinline__ float v8f_get(v8f v, int c) {
  switch (c & 7) {
    case 0: return v[0];
    case 1: return v[1];
    case 2: return v[2];
    case 3: return v[3];
    case 4: return v[4];
    case 5: return v[5];
    case 6: return v[6];
    default: return v[7];
  }
}

__device__ __forceinline__ float log_sigmoid_f(float x) {
  // numerically stable: min(x,0) - log1p(exp(-|x|))
  return fminf(x, 0.0f) - log1pf(expf(-fabsf(x)));
}

__global__ __launch_bounds__(32) void w2v_tile_kernel(
    const int* __restrict__ center_idx,
    const int* __restrict__ context_idx,
    const int* __restrict__ neg_samples,
    const float* __restrict__ in_emb,
    const float* __restrict__ w_out,
    const float* __restrict__ b_out,
    float* __restrict__ partial)
{
  const int lane = threadIdx.x;                 // 0..31, wave32
  const int r0   = blockIdx.x * ROWS_PER_BLOCK; // first batch row of tile
  const int mn   = lane & 15;                   // A-row / B-column for this lane
  const int kh   = (lane >> 4) << 1;            // K sub-offset within 4-chunk

  // Gathered center embedding row for A fragments (A[m][k] = center_m[k]).
  const float* arow = in_emb + (size_t)center_idx[r0 + mn] * W2V_EMBED;

  // Per-column sample index for each of the 6 slots (column n == batch row n).
  int sidx[W2V_SLOTS];
  sidx[0] = context_idx[r0 + mn];
#pragma unroll
  for (int j = 1; j < W2V_SLOTS; ++j)
    sidx[j] = neg_samples[(r0 + mn) * W2V_NEG + (j - 1)];

  const float* brow[W2V_SLOTS];
  float bias[W2V_SLOTS];
#pragma unroll
  for (int j = 0; j < W2V_SLOTS; ++j) {
    brow[j] = w_out + (size_t)sidx[j] * W2V_EMBED;
    bias[j] = b_out[sidx[j]];
  }

  v8f acc[W2V_SLOTS];
#pragma unroll
  for (int j = 0; j < W2V_SLOTS; ++j) acc[j] = (v8f){};

  // K loop: 128 in steps of 4; A loaded once per step, reused by all 6 WMMAs.
  for (int kk = 0; kk < W2V_EMBED; kk += 4) {
    v2f a;
    a.x = arow[kk + kh];
    a.y = arow[kk + kh + 1];
#pragma unroll
    for (int j = 0; j < W2V_SLOTS; ++j) {
      v2f bf;
      bf.x = brow[j][kk + kh];
      bf.y = brow[j][kk + kh + 1];
      // D = A x B + C ; 8 args: (neg_a, A, neg_b, B, c_mod, C, reuse_a, reuse_b)
      acc[j] = __builtin_amdgcn_wmma_f32_16x16x4_f32(
          false, a, false, bf, (short)0, acc[j], false, false);
    }
  }

  // Diagonal extraction per ISA C/D layout:
  //   lanes 0-7  hold D[m][m] for m = lane      at component lane
  //   lanes 24-31 hold D[m][m] for m = lane-16  at component lane-24
  float val = 0.0f;
  const bool has = (lane < 8) || (lane >= 24);
  if (has) {
    const int comp = (lane < 8) ? lane : (lane - 24);
    float s0 = v8f_get(acc[0], comp) + bias[0];
    val = log_sigmoid_f(s0);                      // positive term
#pragma unroll
    for (int j = 1; j < W2V_SLOTS; ++j) {
      float sj = v8f_get(acc[j], comp) + bias[j];
      val += log_sigmoid_f(-sj);                  // negative terms
    }
  }

  // Wave32 reduction; inactive lanes contribute 0.
#pragma unroll
  for (int off = 16; off >= 1; off >>= 1)
    val += __shfl_xor(val, off, 32);

  if (lane == 0) partial[blockIdx.x] = val;
}

__global__ void w2v_finalize_kernel(const float* __restrict__ partial,
                                    float* __restrict__ out, int n) {
  // Fixed-order sum over 128 partials -> deterministic result.
  float s = 0.0f;
  for (int i = 0; i < n; ++i) s += partial[i];
  out[0] = -s / (float)W2V_BATCH;
}

extern "C" void kernel_launch(void* const* d_in, const int* in_sizes, int n_in,
                              void* d_out, int out_size, void* d_ws, size_t ws_size,
                              hipStream_t stream) {
  (void)in_sizes; (void)n_in; (void)out_size; (void)ws_size;

  const int*   center_idx  = (const int*)d_in[0];
  const int*   context_idx = (const int*)d_in[1];
  const int*   neg_samples = (const int*)d_in[2];
  const float* in_emb      = (const float*)d_in[3];
  const float* w_out       = (const float*)d_in[4];
  const float* b_out       = (const float*)d_in[5];
  float*       out         = (float*)d_out;
  float*       partial     = (float*)d_ws;   // 128 floats of scratch

  const int nblocks = W2V_BATCH / ROWS_PER_BLOCK;  // 128 waves
  w2v_tile_kernel<<<nblocks, 32, 0, stream>>>(
      center_idx, context_idx, neg_samples, in_emb, w_out, b_out, partial);
  w2v_finalize_kernel<<<1, 1, 0, stream>>>(partial, out, nblocks);
}